// WLN_80393197846860
// MI455X (gfx1250) — compile-verified
//
#include <hip/hip_runtime.h>
#include <hip/hip_bf16.h>

typedef float v2f __attribute__((ext_vector_type(2)));
typedef float v8f __attribute__((ext_vector_type(8)));

#define HID_D   256
#define NNODES  10000
#define NEDGES  320000
#define NCAND   100000
#define NMP     3
#define NB      4          // N-blocking: each wave computes a 16x64 output strip

// ---------------------------------------------------------------- utilities
__global__ void k_zero(float* __restrict__ p, int n) {
  int i = blockIdx.x * blockDim.x + threadIdx.x;
  if (i < n) p[i] = 0.0f;
}

__global__ void k_init_scores(float* __restrict__ out, const float* __restrict__ b2, int n) {
  int i = blockIdx.x * blockDim.x + threadIdx.x;
  if (i < n) out[i] = b2[0];
}

__global__ void k_degrees(const int* __restrict__ src, const int* __restrict__ dst,
                          float* __restrict__ dego, float* __restrict__ degi, int E) {
  int i = blockIdx.x * blockDim.x + threadIdx.x;
  if (i < E) {
    atomicAdd(&dego[src[i]], 1.0f);
    atomicAdd(&degi[dst[i]], 1.0f);
  }
}

__global__ void k_norms(const float* __restrict__ dego, const float* __restrict__ degi,
                        float* __restrict__ ns, float* __restrict__ nd, int n) {
  int i = blockIdx.x * blockDim.x + threadIdx.x;
  if (i < n) {
    float a = dego[i], b = degi[i];
    ns[i] = (a > 0.0f) ? (1.0f / sqrtf(a)) : 0.0f;
    nd[i] = (b > 0.0f) ? (1.0f / sqrtf(b)) : 0.0f;
  }
}

// message scatter: agg[dst[e]][t] += h[src[e]][t] * norm_src[src[e]]
// one 256-thread block per edge: coalesced gather + coalesced f32 atomics (L2-resident)
__global__ void k_scatter(const float* __restrict__ h, const float* __restrict__ ns,
                          const int* __restrict__ src, const int* __restrict__ dst,
                          float* __restrict__ agg) {
  int e = blockIdx.x;
  int t = threadIdx.x;
  int s  = src[e];
  int dd = dst[e];
  float v = h[s * HID_D + t] * ns[s];
  atomicAdd(&agg[dd * HID_D + t], v);
}

// ------------------------------------------------------------ WMMA f32 GEMM
// out[M x 256] = act( rowScale[r] * A[gather?gather[r]:r][0..K) @ W[K x 256] + bias )
// One wave per 16x64 strip (4 accumulators): A fragment + row scale amortized
// over 4 WMMAs per k-step. Fragment layout per CDNA5 ISA 7.12.2:
//   A  v2f: a[v] = A[M = m0+l16][k0 + 2*half + v]
//   B  v2f: b[v] = W[k0 + 2*half + v][n0 + l16]
//   D  v8f: d[v] = C[m0 + v + 8*half][n0 + l16]
__global__ void k_gemm16(const float* __restrict__ A, const int* __restrict__ gather,
                         const float* __restrict__ rowScale,
                         const float* __restrict__ W, const float* __restrict__ bias,
                         float* __restrict__ out, int M, int K, int relu_en) {
  const int N = HID_D;
  int wave = (blockIdx.x * blockDim.x + threadIdx.x) >> 5;
  int ntn = N / (16 * NB);            // wave-columns across N
  int total = (M / 16) * ntn;
  if (wave >= total) return;          // wave-uniform: EXEC stays all-ones for WMMA
  int tm = wave / ntn, tn = wave % ntn;
  int lane = threadIdx.x & 31;
  int half = lane >> 4, l16 = lane & 15;
  int row = tm * 16 + l16;
  int arow = gather ? gather[row] : row;
  const float* __restrict__ Ar = A + (size_t)arow * K;
  float scl = rowScale ? rowScale[row] : 1.0f;
  int col0 = tn * (16 * NB) + l16;

  v8f acc[NB];
#pragma unroll
  for (int j = 0; j < NB; j++) acc[j] = (v8f){};

  for (int k0 = 0; k0 < K; k0 += 4) {
    int ka = k0 + 2 * half;
    v2f a;
    a.x = Ar[ka] * scl;
    a.y = Ar[ka + 1] * scl;
    const float* __restrict__ Wr0 = W + (size_t)ka * N + col0;
    const float* __restrict__ Wr1 = Wr0 + N;
#pragma unroll
    for (int j = 0; j < NB; j++) {
      v2f b;
      b.x = Wr0[j * 16];
      b.y = Wr1[j * 16];
      acc[j] = __builtin_amdgcn_wmma_f32_16x16x4_f32(false, a, false, b, (short)0,
                                                     acc[j], false, false);
    }
  }
#pragma unroll
  for (int j = 0; j < NB; j++) {
    int col = col0 + j * 16;
    float bb = bias[col];
#pragma unroll
    for (int vv = 0; vv < 8; vv++) {
      float val = acc[j][vv] + bb;
      if (relu_en) val = fmaxf(val, 0.0f);
      out[(size_t)(tm * 16 + 8 * half + vv) * N + col] = val;
    }
  }
}

// ---------------------------------------- fused pair-GEMM + ReLU + GEMV scorer
// hid = relu(concat(h[u],h[v]) @ W1 + b1); scores += hid @ W2 (scores pre-set to b2)
// Wave computes 16x64 of hid, folds its W2 slice in-register, 4 partial atomics
// per 16-row strip instead of 16.
__global__ void k_score(const float* __restrict__ h, const int* __restrict__ cand,
                        const float* __restrict__ W1, const float* __restrict__ b1,
                        const float* __restrict__ W2, float* __restrict__ scores, int M) {
  const int N = HID_D, K = 2 * HID_D;
  int wave = (blockIdx.x * blockDim.x + threadIdx.x) >> 5;
  int ntn = N / (16 * NB);
  int total = (M / 16) * ntn;
  if (wave >= total) return;
  int tm = wave / ntn, tn = wave % ntn;
  int lane = threadIdx.x & 31;
  int half = lane >> 4, l16 = lane & 15;
  int row = tm * 16 + l16;
  int u = cand[2 * row];
  int v = cand[2 * row + 1];
  const float* __restrict__ hu = h + (size_t)u * HID_D;
  const float* __restrict__ hv = h + (size_t)v * HID_D;
  int col0 = tn * (16 * NB) + l16;

  v8f acc[NB];
#pragma unroll
  for (int j = 0; j < NB; j++) acc[j] = (v8f){};

  for (int k0 = 0; k0 < K; k0 += 4) {
    int ka = k0 + 2 * half;
    const float* Ar = (k0 < HID_D) ? (hu + ka) : (hv + (ka - HID_D));
    v2f a;
    a.x = Ar[0];
    a.y = Ar[1];
    const float* __restrict__ Wr0 = W1 + (size_t)ka * N + col0;
    const float* __restrict__ Wr1 = Wr0 + N;
#pragma unroll
    for (int j = 0; j < NB; j++) {
      v2f b;
      b.x = Wr0[j * 16];
      b.y = Wr1[j * 16];
      acc[j] = __builtin_amdgcn_wmma_f32_16x16x4_f32(false, a, false, b, (short)0,
                                                     acc[j], false, false);
    }
  }

  float psum[8];
#pragma unroll
  for (int vv = 0; vv < 8; vv++) psum[vv] = 0.0f;
#pragma unroll
  for (int j = 0; j < NB; j++) {
    int col = col0 + j * 16;
    float bb = b1[col];
    float w2 = W2[col];
#pragma unroll
    for (int vv = 0; vv < 8; vv++)
      psum[vv] += fmaxf(acc[j][vv] + bb, 0.0f) * w2;
  }
  // reduce across the 16 columns of each tile (lanes 0-15 / 16-31 are independent groups)
#pragma unroll
  for (int off = 1; off < 16; off <<= 1) {
#pragma unroll
    for (int vv = 0; vv < 8; vv++)
      psum[vv] += __shfl_xor(psum[vv], off, 32);
  }
  if (l16 == 0) {
#pragma unroll
    for (int vv = 0; vv < 8; vv++)
      atomicAdd(&scores[tm * 16 + 8 * half + vv], psum[vv]);
  }
}

// ------------------------------------------------------------------- launch
extern "C" void kernel_launch(void* const* d_in, const int* in_sizes, int n_in,
                              void* d_out, int out_size, void* d_ws, size_t ws_size,
                              hipStream_t stream) {
  const int*   atom_types = (const int*)d_in[0];
  const int*   edge_index = (const int*)d_in[2];   // [0][:] src, [1][:] dst
  const int*   cand       = (const int*)d_in[3];   // (NCAND, 2) row-major
  const float* atom_embed = (const float*)d_in[4];
  const float* atom_W     = (const float*)d_in[5];
  const float* atom_b     = (const float*)d_in[6];
  const float* conv_W     = (const float*)d_in[10]; // (3, 256, 256)
  const float* conv_b     = (const float*)d_in[11]; // (3, 256)
  const float* s_W1       = (const float*)d_in[12]; // (512, 256)
  const float* s_b1       = (const float*)d_in[13];
  const float* s_W2       = (const float*)d_in[14]; // (256, 1)
  const float* s_b2       = (const float*)d_in[15];
  float* scores = (float*)d_out;

  const int* e_src = edge_index;
  const int* e_dst = edge_index + NEDGES;

  // workspace layout (floats): h | agg | deg_out | deg_in | norm_src | norm_dst
  float* h    = (float*)d_ws;
  float* agg  = h + (size_t)NNODES * HID_D;
  float* dego = agg + (size_t)NNODES * HID_D;
  float* degi = dego + NNODES;
  float* ns   = degi + NNODES;
  float* nd   = ns + NNODES;

  // degrees + normalizers
  k_zero<<<(2 * NNODES + 255) / 256, 256, 0, stream>>>(dego, 2 * NNODES);
  k_degrees<<<(NEDGES + 255) / 256, 256, 0, stream>>>(e_src, e_dst, dego, degi, NEDGES);
  k_norms<<<(NNODES + 255) / 256, 256, 0, stream>>>(dego, degi, ns, nd, NNODES);

  // h = atom_embed[atom_types] @ atom_W + atom_b   (no activation)
  {
    int tiles = (NNODES / 16) * (HID_D / (16 * NB)); // 2500 waves
    int blocks = (tiles + 7) / 8;                    // 8 waves / 256-thread block
    k_gemm16<<<blocks, 256, 0, stream>>>(atom_embed, atom_types, nullptr,
                                         atom_W, atom_b, h, NNODES, HID_D, 0);
  }

  // 3 message-passing layers
  for (int i = 0; i < NMP; i++) {
    k_zero<<<(NNODES * HID_D + 255) / 256, 256, 0, stream>>>(agg, NNODES * HID_D);
    k_scatter<<<NEDGES, HID_D, 0, stream>>>(h, ns, e_src, e_dst, agg);
    int tiles = (NNODES / 16) * (HID_D / (16 * NB));
    int blocks = (tiles + 7) / 8;
    k_gemm16<<<blocks, 256, 0, stream>>>(agg, nullptr, nd,
                                         conv_W + (size_t)i * HID_D * HID_D,
                                         conv_b + (size_t)i * HID_D,
                                         h, NNODES, HID_D, 1);
  }

  // scorer: scores = relu(concat(h[u],h[v]) @ W1 + b1) @ W2 + b2
  k_init_scores<<<(NCAND + 255) / 256, 256, 0, stream>>>(scores, s_b2, NCAND);
  {
    int tiles = (NCAND / 16) * (HID_D / (16 * NB));  // 25000 waves
    int blocks = (tiles + 7) / 8;
    k_score<<<blocks, 256, 0, stream>>>(h, cand, s_W1, s_b1, s_W2, scores, NCAND);
  }
}